// CrossAttn_5763846111589
// MI455X (gfx1250) — compile-verified
//
#include <hip/hip_runtime.h>
#include <math.h>

typedef __attribute__((ext_vector_type(2))) float v2f;
typedef __attribute__((ext_vector_type(8))) float v8f;

#define NQ   32768
#define MR   8192
#define CCH  128
#define KNN  8

// ============================================================================
// Generic Y[rows x 128] = X[rows x 128] @ W^T + b, f32 WMMA 16x16x4.
// Block = 128 threads (4 waves). Each wave computes a 16-row stripe.
// ============================================================================
#define GW_STRIDE 132   // 132 words: 16B-aligned float4 staging, conflict-free b64 frags

__global__ __launch_bounds__(128)
void gemm128_wmma(const float* __restrict__ X, const float* __restrict__ W,
                  const float* __restrict__ B, float* __restrict__ Y) {
  __shared__ alignas(16) float wt[16 * GW_STRIDE];        // one 16-col tile of W
  __shared__ alignas(16) float xt[4][16 * GW_STRIDE];     // per-wave X tile
  __shared__ float bs[CCH];

  const int tid  = threadIdx.x;
  const int wave = tid >> 5;
  const int lane = tid & 31;
  const int l15  = lane & 15;
  const int hi   = lane >> 4;                 // selects K pair (A/B) / row half (D)
  const long row0 = (long)blockIdx.x * 64 + wave * 16;

  // Stage this wave's 16x128 X tile, coalesced float4 per row.
  #pragma unroll 4
  for (int r = 0; r < 16; ++r) {
    float4 v = *(const float4*)(X + (row0 + r) * CCH + lane * 4);
    *(float4*)&xt[wave][r * GW_STRIDE + lane * 4] = v;
  }
  bs[tid] = B[tid];
  __syncthreads();

  const float* xw = &xt[wave][0];

  for (int t = 0; t < 8; ++t) {               // 8 column tiles of 16
    // Cooperatively stage W rows [t*16, t*16+16) : 512 float4, 4 per thread.
    #pragma unroll
    for (int i = 0; i < 4; ++i) {
      int fi = tid + i * 128;
      int r  = fi >> 5;
      int c  = (fi & 31) * 4;
      *(float4*)&wt[r * GW_STRIDE + c] = *(const float4*)(W + (t * 16 + r) * CCH + c);
    }
    __syncthreads();

    float fb = bs[t * 16 + l15];
    v8f acc = {fb, fb, fb, fb, fb, fb, fb, fb};   // bias broadcast down column n

    #pragma unroll 8
    for (int kk = 0; kk < CCH; kk += 4) {
      v2f a = *(const v2f*)&xw[l15 * GW_STRIDE + kk + 2 * hi];  // A[m=l15][kk+2hi ..]
      v2f b = *(const v2f*)&wt[l15 * GW_STRIDE + kk + 2 * hi];  // B[k][n=l15] = W[n][k]
      acc = __builtin_amdgcn_wmma_f32_16x16x4_f32(
          false, a, false, b, (short)0, acc, false, false);
    }

    #pragma unroll
    for (int r = 0; r < 8; ++r)
      Y[(row0 + r + 8 * hi) * CCH + t * 16 + l15] = acc[r];

    __syncthreads();   // protect wt before next tile's staging
  }
}

// ============================================================================
// KNN: per wave, 16 queries vs all 8192 refs via WMMA score tiles.
// Score d' = |r|^2 - 2 q.r  (same ranking as true sq-distance).
// A rows = refs (-2x,-2y,-2z,|r|^2), B cols = queries (x,y,z,1), K=4.
// D: lane l holds query (l&15), refs r + 8*(l>=16)  -> per-lane top-8.
// ============================================================================
#define KCHUNK 2048   // refs staged per LDS pass (32 KB)

__global__ __launch_bounds__(256)
void knn_kernel(const float* __restrict__ xyz_pred,
                const float* __restrict__ xyz_ref,
                int* __restrict__ out_idx) {
  __shared__ alignas(16) float4 refpack[KCHUNK];
  __shared__ float mdist[8][16][16];
  __shared__ int   midx [8][16][16];

  const int tid  = threadIdx.x;
  const int wave = tid >> 5;
  const int lane = tid & 31;
  const int l15  = lane & 15;
  const int hi   = lane >> 4;
  const int q    = blockIdx.x * 128 + wave * 16 + l15;

  // B fragment (constant all loop): lanes 0-15 -> (qx,qy); lanes 16-31 -> (qz,1)
  v2f bq;
  if (hi == 0) {
    bq.x = xyz_pred[q * 3 + 0];
    bq.y = xyz_pred[q * 3 + 1];
  } else {
    bq.x = xyz_pred[q * 3 + 2];
    bq.y = 1.0f;
  }

  float dd[KNN];
  int   ii[KNN];
  #pragma unroll
  for (int s = 0; s < KNN; ++s) { dd[s] = 3.0e38f; ii[s] = 0; }

  for (int ch = 0; ch < MR; ch += KCHUNK) {
    __syncthreads();
    for (int m = tid; m < KCHUNK; m += 256) {
      float x = xyz_ref[(ch + m) * 3 + 0];
      float y = xyz_ref[(ch + m) * 3 + 1];
      float z = xyz_ref[(ch + m) * 3 + 2];
      refpack[m] = make_float4(-2.0f * x, -2.0f * y, -2.0f * z,
                               x * x + y * y + z * z);
    }
    __syncthreads();

    for (int t = 0; t < KCHUNK / 16; ++t) {
      const float* rp = (const float*)&refpack[t * 16 + l15];
      v2f a = *(const v2f*)(rp + 2 * hi);          // (-2x,-2y) or (-2z,|r|^2)
      v8f cz = {0.f, 0.f, 0.f, 0.f, 0.f, 0.f, 0.f, 0.f};
      v8f d = __builtin_amdgcn_wmma_f32_16x16x4_f32(
          false, a, false, bq, (short)0, cz, false, false);

      const int mbase = ch + t * 16 + 8 * hi;
      #pragma unroll
      for (int r = 0; r < 8; ++r) {
        float dv = d[r];
        if (dv < dd[KNN - 1]) {                    // rare (~1%) insert path
          float cd = dv;
          int   ci = mbase + r;
          #pragma unroll
          for (int s = 0; s < KNN; ++s) {          // branchless swap-down insert
            bool  lt = cd < dd[s];
            float td = dd[s]; int ti = ii[s];
            dd[s] = lt ? cd : td;  ii[s] = lt ? ci : ti;
            cd    = lt ? td : cd;  ci    = lt ? ti : ci;
          }
        }
      }
    }
  }

  // Merge the two sorted 8-lists of a lane pair (refs interleaved by 8).
  #pragma unroll
  for (int s = 0; s < KNN; ++s) {
    mdist[wave][l15][hi * 8 + s] = dd[s];
    midx [wave][l15][hi * 8 + s] = ii[s];
  }
  __syncthreads();
  if (lane < 16) {
    int i = 0, j = 0;
    #pragma unroll
    for (int s = 0; s < KNN; ++s) {
      float da = mdist[wave][l15][i];
      float db = mdist[wave][l15][8 + j];
      bool  ta = da <= db;
      out_idx[q * KNN + s] = ta ? midx[wave][l15][i] : midx[wave][l15][8 + j];
      i += ta ? 1 : 0;
      j += ta ? 0 : 1;
    }
  }
}

// ============================================================================
// Per-query 8-neighbor attention. One wave per query; lane = 4 channels.
// ============================================================================
__global__ __launch_bounds__(256)
void attn_kernel(const float* __restrict__ q_feat,
                 const float* __restrict__ k_feat,
                 const float* __restrict__ v_raw,
                 const int* __restrict__ knn,
                 float* __restrict__ pred) {
  const int wave = threadIdx.x >> 5;
  const int lane = threadIdx.x & 31;
  const long nq  = (long)blockIdx.x * 8 + wave;
  const int  c0  = lane * 4;

  const float4 qv = *(const float4*)(q_feat + nq * CCH + c0);

  int id[KNN];
  #pragma unroll
  for (int j = 0; j < KNN; ++j) id[j] = knn[nq * KNN + j];

  float p[KNN];
  #pragma unroll
  for (int j = 0; j < KNN; ++j) {
    const float4 kv = *(const float4*)(k_feat + (long)id[j] * CCH + c0);
    p[j] = qv.x * kv.x + qv.y * kv.y + qv.z * kv.z + qv.w * kv.w;
  }
  #pragma unroll
  for (int off = 16; off > 0; off >>= 1) {
    #pragma unroll
    for (int j = 0; j < KNN; ++j) p[j] += __shfl_xor(p[j], off, 32);
  }

  const float scale = 0.08838834764831845f;   // 1/sqrt(128)
  float mx = -3.0e38f;
  #pragma unroll
  for (int j = 0; j < KNN; ++j) { p[j] *= scale; mx = fmaxf(mx, p[j]); }
  float sum = 0.0f, e[KNN];
  #pragma unroll
  for (int j = 0; j < KNN; ++j) { e[j] = __expf(p[j] - mx); sum += e[j]; }
  const float inv = 1.0f / sum;

  float4 o = {0.f, 0.f, 0.f, 0.f};
  #pragma unroll
  for (int j = 0; j < KNN; ++j) {
    float a = e[j] * inv;
    const float4 vv = *(const float4*)(v_raw + (long)id[j] * CCH + c0);
    o.x += a * vv.x; o.y += a * vv.y; o.z += a * vv.z; o.w += a * vv.w;
  }
  *(float4*)(pred + nq * CCH + c0) = o;
}

// ============================================================================
// Launch: vraw = vfeat@Wv^T+bv | knn idx | pred(->d_out) | h = pred@Wo^T+bo
//         out = h@Wout^T+bout  (workspace: 4MB vraw + 1MB idx + 16MB h)
// ============================================================================
extern "C" void kernel_launch(void* const* d_in, const int* in_sizes, int n_in,
                              void* d_out, int out_size, void* d_ws, size_t ws_size,
                              hipStream_t stream) {
  const float* xyz_pred = (const float*)d_in[0];
  const float* xyz_ref  = (const float*)d_in[1];
  const float* q_feat   = (const float*)d_in[2];
  const float* k_feat   = (const float*)d_in[3];
  const float* v_feat   = (const float*)d_in[4];
  const float* W_v   = (const float*)d_in[5];
  const float* b_v   = (const float*)d_in[6];
  const float* W_o   = (const float*)d_in[7];
  const float* b_o   = (const float*)d_in[8];
  const float* W_out = (const float*)d_in[9];
  const float* b_out = (const float*)d_in[10];
  float* out = (float*)d_out;

  char*  ws   = (char*)d_ws;
  float* vraw = (float*)ws;                                          // 4 MiB
  int*   kidx = (int*)(ws + (size_t)MR * CCH * 4);                   // 1 MiB
  float* hbuf = (float*)(ws + (size_t)MR * CCH * 4
                            + (size_t)NQ * KNN * 4);                 // 16 MiB

  gemm128_wmma<<<MR / 64, 128, 0, stream>>>(v_feat, W_v, b_v, vraw);
  knn_kernel  <<<NQ / 128, 256, 0, stream>>>(xyz_pred, xyz_ref, kidx);
  attn_kernel <<<NQ / 8, 256, 0, stream>>>(q_feat, k_feat, vraw, kidx, out);
  gemm128_wmma<<<NQ / 64, 128, 0, stream>>>(out, W_o, b_o, hbuf);
  gemm128_wmma<<<NQ / 64, 128, 0, stream>>>(hbuf, W_out, b_out, out);
}